// LowRankModl_30889404793085
// MI455X (gfx1250) — compile-verified
//
#include <hip/hip_runtime.h>
#include <math.h>

// ---------------------------------------------------------------------------
// Problem constants (from reference): B=1, T=12, C=10, H=W=256, SV=3
// ---------------------------------------------------------------------------
#define TT      12
#define CC      10
#define HH      256
#define WW      256
#define SV      3
#define NPIX    (HH * WW)          // 65536
#define NVEC    (SV * NPIX)        // 196608 complex elements of a CG vector
#define LAM     0.05f
#define CG_ITERS 10

typedef float v2f __attribute__((ext_vector_type(2)));
typedef float v8f __attribute__((ext_vector_type(8)));

// ------------------------- complex helpers ---------------------------------
__device__ __forceinline__ float2 cmul(float2 a, float2 b) {
    return make_float2(a.x * b.x - a.y * b.y, a.x * b.y + a.y * b.x);
}
__device__ __forceinline__ float2 cmulc(float2 a, float2 b) {  // conj(a)*b
    return make_float2(a.x * b.x + a.y * b.y, a.x * b.y - a.y * b.x);
}
__device__ __forceinline__ float2 cadd(float2 a, float2 b) {
    return make_float2(a.x + b.x, a.y + b.y);
}

// f32 WMMA 16x16x4:  D = A(16x4) * B(4x16) + C  (one 16x16 tile per wave32)
__device__ __forceinline__ v8f wmma4(v2f a, v2f b, v8f c) {
    return __builtin_amdgcn_wmma_f32_16x16x4_f32(false, a, false, b,
                                                 (short)0, c, false, false);
}

// Intra-wave LDS pipeline fence: same-wave DS ops are architecturally
// in-order on CDNA5, so a compiler code-motion barrier is sufficient —
// no s_barrier needed (each wave owns a private tile slice).
__device__ __forceinline__ void wave_fence() {
    __builtin_amdgcn_wave_barrier();
}

// ---------------------------------------------------------------------------
// Batched 256-point transforms along rows (fastest dim), transposed output.
// One wave = one row transform via four-step: X = W16 * (tw .* (W16 * M))^T,
// every 16x16 complex GEMM = 16x V_WMMA_F32_16X16X4_F32.
//   dir  : -1 -> fft2 convention exp(-2*pi*i*n*k/N),  +1 -> ifft (no 1/N)
//   scale: applied per 1D pass (folded into twiddles; ifft pass uses 1/256)
//   chi  : multiply input by (-1)^(row+col)   (centered-FFT shift folding)
//   mulin: optional pointwise complex multiplier on input (natural layout)
//   mulout: if non-null, epilogue  out[img] += conj(mulout) * result
// Grid: (4, nimg), block 256 (8 waves, 8 rows each, fully wave-independent).
// ---------------------------------------------------------------------------
__global__ __launch_bounds__(256)
void fft256_kernel(const float2* __restrict__ in, float2* __restrict__ out,
                   const float2* __restrict__ mulin, int chi,
                   const float2* __restrict__ mulout,
                   float dir, float scale) {
    __shared__ float2 tile[8][264];

    const int lane = threadIdx.x & 31;
    const int wv   = threadIdx.x >> 5;
    const int half = lane >> 4;
    const int r    = lane & 15;      // output-row of the 16x16 tile this lane owns
    const int img  = blockIdx.y;

    // Constant 16-pt DFT matrix operands (A layout: lane row = r,
    // chunk kk covers K-cols 4kk..4kk+3; reg j -> col 4kk + 2*half + j).
    v2f Wr[4], Wi[4], WiN[4];
#pragma unroll
    for (int kk = 0; kk < 4; ++kk) {
#pragma unroll
        for (int j = 0; j < 2; ++j) {
            int   col = kk * 4 + half * 2 + j;
            float th  = dir * 0.39269908169872414f * (float)(r * col); // 2pi/16
            float s, c;
            __sincosf(th, &s, &c);
            Wr[kk][j]  = c;
            Wi[kk][j]  = s;
            WiN[kk][j] = -s;
        }
    }

    // Loop-invariant twiddles tw[k1][n2=r] with the output scale folded in.
    float twc[8], tws[8];
#pragma unroll
    for (int j = 0; j < 8; ++j) {
        int   k1 = half * 8 + j;
        float th = dir * 0.024543692606170259f * (float)(k1 * r);   // 2pi/256
        float s, c;
        __sincosf(th, &s, &c);
        twc[j] = c * scale;
        tws[j] = s * scale;
    }

    const float2* inImg = in + (size_t)img * NPIX;

    for (int rep = 0; rep < 8; ++rep) {
        const int row = blockIdx.x * 64 + wv * 8 + rep;

        // ---- load one row (256 complex) into the wave's LDS tile ----------
#pragma unroll
        for (int q = 0; q < 8; ++q) {
            int    idx = lane * 8 + q;
            float2 v   = inImg[(size_t)row * WW + idx];
            if (chi && ((row + idx) & 1)) { v.x = -v.x; v.y = -v.y; }
            if (mulin) v = cmul(v, mulin[(size_t)row * WW + idx]);
            tile[wv][idx] = v;
        }
        if (rep < 7)
            __builtin_prefetch(&inImg[(size_t)(row + 1) * WW + lane * 8], 0, 1);
        wave_fence();

        // ---- step 1: T = W16 * M  (M[n1][n2] = tile[n1*16+n2]) ------------
        v8f Tr = {0.f, 0.f, 0.f, 0.f, 0.f, 0.f, 0.f, 0.f};
        v8f Ti = {0.f, 0.f, 0.f, 0.f, 0.f, 0.f, 0.f, 0.f};
#pragma unroll
        for (int kk = 0; kk < 4; ++kk) {
            int    rowB = kk * 4 + half * 2;
            float2 e0   = tile[wv][(rowB + 0) * 16 + r];
            float2 e1   = tile[wv][(rowB + 1) * 16 + r];
            v2f br, bi;
            br[0] = e0.x; br[1] = e1.x;
            bi[0] = e0.y; bi[1] = e1.y;
            Tr = wmma4(Wr[kk],  br, Tr);
            Tr = wmma4(WiN[kk], bi, Tr);
            Ti = wmma4(Wr[kk],  bi, Ti);
            Ti = wmma4(Wi[kk],  br, Ti);
        }
        wave_fence();

        // ---- step 2: twiddle (+scale), write T' back to LDS ---------------
#pragma unroll
        for (int j = 0; j < 8; ++j) {
            int   k1 = half * 8 + j;
            float tr = Tr[j], ti = Ti[j];
            tile[wv][k1 * 16 + r] =
                make_float2(tr * twc[j] - ti * tws[j], tr * tws[j] + ti * twc[j]);
        }
        wave_fence();

        // ---- step 3: R = W16 * T'^T  (B[n2][k1] = T'[k1][n2]) -------------
        v8f Rr = {0.f, 0.f, 0.f, 0.f, 0.f, 0.f, 0.f, 0.f};
        v8f Ri = {0.f, 0.f, 0.f, 0.f, 0.f, 0.f, 0.f, 0.f};
#pragma unroll
        for (int kk = 0; kk < 4; ++kk) {
            int    n2b = kk * 4 + half * 2;
            float2 e0  = tile[wv][r * 16 + n2b + 0];
            float2 e1  = tile[wv][r * 16 + n2b + 1];
            v2f br, bi;
            br[0] = e0.x; br[1] = e1.x;
            bi[0] = e0.y; bi[1] = e1.y;
            Rr = wmma4(Wr[kk],  br, Rr);
            Rr = wmma4(WiN[kk], bi, Rr);
            Ri = wmma4(Wr[kk],  bi, Ri);
            Ri = wmma4(Wi[kk],  br, Ri);
        }
        wave_fence();

        // ---- pack result vector X[k2*16+k1] into LDS ----------------------
#pragma unroll
        for (int j = 0; j < 8; ++j) {
            int k2 = half * 8 + j;
            tile[wv][k2 * 16 + r] = make_float2(Rr[j], Ri[j]);
        }
        wave_fence();

        // ---- transposed store (optionally fused conj(mulout)*res += out) --
        float2* outImg = out + (size_t)img * NPIX;
#pragma unroll
        for (int q = 0; q < 8; ++q) {
            int    idx = lane * 8 + q;
            float2 v   = tile[wv][idx];
            size_t op  = (size_t)idx * HH + row;   // transposed pixel
            if (mulout) {
                float2 res = cmulc(mulout[op], v);
                outImg[op].x += res.x;
                outImg[op].y += res.y;
            } else {
                outImg[op] = v;
            }
        }
        wave_fence();
    }
}

// ---------------------------------------------------------------------------
// Pointwise / reduction kernels
// ---------------------------------------------------------------------------
__global__ void prep_stilde(const float2* __restrict__ sens,
                            float2* __restrict__ st) {
    int i = blockIdx.x * blockDim.x + threadIdx.x;      // CC*NPIX threads
    int pix = i & (NPIX - 1);
    int rr = pix >> 8, cc = pix & 255;
    float sgn = ((rr + cc) & 1) ? -1.f : 1.f;
    float2 v = sens[i];
    st[i] = make_float2(v.x * sgn, v.y * sgn);
}

// K[s][s'](k) = sum_t conj(L[t,s]) * L[t,s'] * mask[t](k)
__global__ void prep_K(const float2* __restrict__ L,
                       const float* __restrict__ mask,
                       float2* __restrict__ K) {
    int pix = blockIdx.x * blockDim.x + threadIdx.x;    // NPIX threads
    float2 acc[SV][SV];
#pragma unroll
    for (int s = 0; s < SV; ++s)
#pragma unroll
        for (int t = 0; t < SV; ++t) acc[s][t] = make_float2(0.f, 0.f);
    for (int t = 0; t < TT; ++t) {
        float m = mask[(size_t)t * NPIX + pix];
        if (m != 0.f) {
#pragma unroll
            for (int s = 0; s < SV; ++s) {
                float2 ls = L[t * SV + s];
#pragma unroll
                for (int sp = 0; sp < SV; ++sp) {
                    float2 lp = L[t * SV + sp];
                    float2 pr = cmulc(ls, lp);
                    acc[s][sp].x += m * pr.x;
                    acc[s][sp].y += m * pr.y;
                }
            }
        }
    }
#pragma unroll
    for (int s = 0; s < SV; ++s)
#pragma unroll
        for (int sp = 0; sp < SV; ++sp)
            K[(size_t)(s * SV + sp) * NPIX + pix] = acc[s][sp];
}

// ahy_t = sum_c conj(Stilde_c) * g_c
__global__ void combine_coils(const float2* __restrict__ g,
                              const float2* __restrict__ st,
                              float2* __restrict__ ahy_t) {
    int pix = blockIdx.x * blockDim.x + threadIdx.x;
    float2 acc = make_float2(0.f, 0.f);
    for (int c = 0; c < CC; ++c)
        acc = cadd(acc, cmulc(st[(size_t)c * NPIX + pix],
                              g[(size_t)c * NPIX + pix]));
    ahy_t[pix] = acc;
}

// b_s = lambda*mo_s + sum_t conj(L[t,s]) * ahy_t
__global__ void bvec_kernel(const float2* __restrict__ L,
                            const float2* __restrict__ ahy,
                            const float2* __restrict__ mo,
                            float2* __restrict__ bv) {
    int pix = blockIdx.x * blockDim.x + threadIdx.x;
#pragma unroll
    for (int s = 0; s < SV; ++s) {
        float2 m = mo[(size_t)s * NPIX + pix];
        float2 acc = make_float2(LAM * m.x, LAM * m.y);
        for (int t = 0; t < TT; ++t)
            acc = cadd(acc, cmulc(L[t * SV + s], ahy[(size_t)t * NPIX + pix]));
        bv[(size_t)s * NPIX + pix] = acc;
    }
}

// v_s = sum_s' K[s][s'] * u_s'
__global__ void kmix_kernel(const float2* __restrict__ u,
                            const float2* __restrict__ K,
                            float2* __restrict__ v) {
    int pix = blockIdx.x * blockDim.x + threadIdx.x;
    float2 uu[SV];
#pragma unroll
    for (int s = 0; s < SV; ++s) uu[s] = u[(size_t)s * NPIX + pix];
#pragma unroll
    for (int s = 0; s < SV; ++s) {
        float2 acc = make_float2(0.f, 0.f);
#pragma unroll
        for (int sp = 0; sp < SV; ++sp)
            acc = cadd(acc, cmul(K[(size_t)(s * SV + sp) * NPIX + pix], uu[sp]));
        v[(size_t)s * NPIX + pix] = acc;
    }
}

__global__ void cg_init(const float2* __restrict__ bv, float2* __restrict__ x,
                        float2* __restrict__ r, float2* __restrict__ p) {
    int i = blockIdx.x * blockDim.x + threadIdx.x;
    float2 b = bv[i];
    x[i] = make_float2(0.f, 0.f);
    r[i] = b;
    p[i] = b;
}

__global__ void init_Ap(const float2* __restrict__ p, float2* __restrict__ Ap) {
    int i = blockIdx.x * blockDim.x + threadIdx.x;
    float2 v = p[i];
    Ap[i] = make_float2(LAM * v.x, LAM * v.y);
}

// part[block] = sum conj(a)*b over a strided range (deterministic tree)
__global__ void dot_partial(const float2* __restrict__ a,
                            const float2* __restrict__ b,
                            float2* __restrict__ part, int n) {
    __shared__ float2 sm[256];
    float2 acc = make_float2(0.f, 0.f);
    for (int i = blockIdx.x * 256 + threadIdx.x; i < n; i += 256 * 256)
        acc = cadd(acc, cmulc(a[i], b[i]));
    sm[threadIdx.x] = acc;
    __syncthreads();
    for (int s = 128; s > 0; s >>= 1) {
        if (threadIdx.x < s)
            sm[threadIdx.x] = cadd(sm[threadIdx.x], sm[threadIdx.x + s]);
        __syncthreads();
    }
    if (threadIdx.x == 0) part[blockIdx.x] = sm[0];
}

__global__ void dot_final(const float2* __restrict__ part,
                          float2* __restrict__ slot) {
    __shared__ float2 sm[256];
    sm[threadIdx.x] = part[threadIdx.x];
    __syncthreads();
    for (int s = 128; s > 0; s >>= 1) {
        if (threadIdx.x < s)
            sm[threadIdx.x] = cadd(sm[threadIdx.x], sm[threadIdx.x + s]);
        __syncthreads();
    }
    if (threadIdx.x == 0) slot[0] = sm[0];
}

// slots[0]=rs_old (real), slots[1]=<p,Ap> (complex), slots[2]=rs_new
__global__ void cg_update_xr(float2* __restrict__ x, float2* __restrict__ r,
                             const float2* __restrict__ p,
                             const float2* __restrict__ Ap,
                             const float2* __restrict__ slots) {
    int i = blockIdx.x * blockDim.x + threadIdx.x;
    float  rs  = slots[0].x;
    float2 pap = slots[1];
    float  den = pap.x * pap.x + pap.y * pap.y;
    float2 alpha = make_float2(rs * pap.x / den, -rs * pap.y / den);
    float2 pi = p[i], api = Ap[i];
    float2 xi = x[i], ri = r[i];
    float2 ap  = cmul(alpha, pi);
    float2 aap = cmul(alpha, api);
    x[i] = make_float2(xi.x + ap.x, xi.y + ap.y);
    r[i] = make_float2(ri.x - aap.x, ri.y - aap.y);
}

__global__ void cg_update_p(float2* __restrict__ p, const float2* __restrict__ r,
                            const float2* __restrict__ slots) {
    int i = blockIdx.x * blockDim.x + threadIdx.x;
    float beta = slots[2].x / slots[0].x;
    float2 ri = r[i], pi = p[i];
    p[i] = make_float2(ri.x + beta * pi.x, ri.y + beta * pi.y);
}

__global__ void scalar_shift(float2* __restrict__ slots) {
    if (threadIdx.x == 0) slots[0] = slots[2];
}

__global__ void out_copy(const float2* __restrict__ x, float2* __restrict__ o) {
    int i = blockIdx.x * blockDim.x + threadIdx.x;
    o[i] = x[i];
}

// ---------------------------------------------------------------------------
// Host orchestration (graph-capturable: all launches on `stream`)
// ---------------------------------------------------------------------------
extern "C" void kernel_launch(void* const* d_in, const int* in_sizes, int n_in,
                              void* d_out, int out_size, void* d_ws, size_t ws_size,
                              hipStream_t stream) {
    (void)in_sizes; (void)n_in; (void)out_size; (void)ws_size;

    const float2* y    = (const float2*)d_in[0];   // [T,C,H,W] complex
    const float2* mo   = (const float2*)d_in[1];   // [SV,H,W]
    const float2* sens = (const float2*)d_in[2];   // [C,H,W]
    const float2* L    = (const float2*)d_in[3];   // [T,SV]
    const float*  mask = (const float*)d_in[4];    // [T,H,W]

    char* w = (char*)d_ws;
    auto carve = [&](size_t nc) { float2* q = (float2*)w; w += nc * sizeof(float2); return q; };
    float2* St    = carve((size_t)CC * NPIX);       // chi-folded sens
    float2* Kmat  = carve((size_t)SV * SV * NPIX);  // spatio-temporal kernel
    float2* ahy   = carve((size_t)TT * NPIX);
    float2* bv    = carve(NVEC);
    float2* x     = carve(NVEC);
    float2* r     = carve(NVEC);
    float2* p     = carve(NVEC);
    float2* Ap    = carve(NVEC);
    float2* bufA  = carve((size_t)CC * NPIX);
    float2* bufB  = carve((size_t)CC * NPIX);
    float2* part  = carve(256);
    float2* slots = carve(8);

    const dim3 blk(256);
    const float INV256 = 1.0f / 256.0f;

    // ---- precompute S~ and K ------------------------------------------------
    prep_stilde<<<(CC * NPIX) / 256, blk, 0, stream>>>(sens, St);
    prep_K<<<NPIX / 256, blk, 0, stream>>>(L, mask, Kmat);

    // ---- b-vector: ahy_t = sum_c conj(S~_c) * fft2(chi * y_tc) -------------
    for (int t = 0; t < TT; ++t) {
        const float2* yt = y + (size_t)t * CC * NPIX;
        fft256_kernel<<<dim3(4, CC), blk, 0, stream>>>(yt, bufA, nullptr, 1,
                                                       nullptr, -1.f, 1.f);
        fft256_kernel<<<dim3(4, CC), blk, 0, stream>>>(bufA, bufB, nullptr, 0,
                                                       nullptr, -1.f, 1.f);
        combine_coils<<<NPIX / 256, blk, 0, stream>>>(bufB, St,
                                                      ahy + (size_t)t * NPIX);
    }
    bvec_kernel<<<NPIX / 256, blk, 0, stream>>>(L, ahy, mo, bv);

    // ---- CG: x0 = 0, r = p = b, rs = <r,r> ---------------------------------
    cg_init<<<NVEC / 256, blk, 0, stream>>>(bv, x, r, p);
    dot_partial<<<256, blk, 0, stream>>>(r, r, part, NVEC);
    dot_final<<<1, blk, 0, stream>>>(part, slots + 0);

    for (int it = 0; it < CG_ITERS; ++it) {
        // Ap = lambda * p, then accumulate coil terms
        init_Ap<<<NVEC / 256, blk, 0, stream>>>(p, Ap);
        for (int c = 0; c < CC; ++c) {
            const float2* Sc = St + (size_t)c * NPIX;
            // ifft2( S~_c * p_s' )   (scale 1/256 per 1D pass)
            fft256_kernel<<<dim3(4, SV), blk, 0, stream>>>(p, bufA, Sc, 0,
                                                           nullptr, +1.f, INV256);
            fft256_kernel<<<dim3(4, SV), blk, 0, stream>>>(bufA, bufB, nullptr, 0,
                                                           nullptr, +1.f, INV256);
            // k-space mixing with K
            kmix_kernel<<<NPIX / 256, blk, 0, stream>>>(bufB, Kmat, bufA);
            // fft2, fused epilogue: Ap_s += conj(S~_c) * result
            fft256_kernel<<<dim3(4, SV), blk, 0, stream>>>(bufA, bufB, nullptr, 0,
                                                           nullptr, -1.f, 1.f);
            fft256_kernel<<<dim3(4, SV), blk, 0, stream>>>(bufB, Ap, nullptr, 0,
                                                           Sc, -1.f, 1.f);
        }
        // alpha = rs / <p,Ap>;  x += alpha p;  r -= alpha Ap
        dot_partial<<<256, blk, 0, stream>>>(p, Ap, part, NVEC);
        dot_final<<<1, blk, 0, stream>>>(part, slots + 1);
        cg_update_xr<<<NVEC / 256, blk, 0, stream>>>(x, r, p, Ap, slots);
        // rs_new = <r,r>;  p = r + (rs_new/rs) p;  rs = rs_new
        dot_partial<<<256, blk, 0, stream>>>(r, r, part, NVEC);
        dot_final<<<1, blk, 0, stream>>>(part, slots + 2);
        cg_update_p<<<NVEC / 256, blk, 0, stream>>>(p, r, slots);
        scalar_shift<<<1, 32, 0, stream>>>(slots);
    }

    out_copy<<<NVEC / 256, blk, 0, stream>>>(x, (float2*)d_out);
}